// TripletLossBatchAll_71820443124407
// MI455X (gfx1250) — compile-verified
//
#include <hip/hip_runtime.h>
#include <hip/hip_bf16.h>

typedef float v2f __attribute__((ext_vector_type(2)));
typedef float v8f __attribute__((ext_vector_type(8)));

#define NPTS 512
#define DIM  128
#define TLMARGIN 0.3f
#define TLEPS 1e-12f

// ---------------------------------------------------------------------------
// Kernel 1: Gram matrix G = X * X^T via V_WMMA_F32_16X16X4_F32.
// One wave32 per 16x16 output tile; 32 chained WMMAs cover K=128.
//
// f32 A (16x4) fragment layout (ISA 7.12.2): lanes 0-15 hold K={0,1} in
// VGPR{0,1}, lanes 16-31 hold K={2,3}.  B (4x16) mirrors this with N in
// place of M, and since B = A^T for a Gram matrix, both fragments load with
// the same pattern: lane loads float2 at X[tileRow*16 + (lane&15)][k0 + 2*hi].
// C/D layout: VGPR v -> (M = v + 8*hi, N = lane&15).
// ---------------------------------------------------------------------------
__global__ __launch_bounds__(128) void gram_wmma_kernel(
    const float* __restrict__ X, float* __restrict__ G) {
  const int wave = threadIdx.x >> 5;
  const int lane = threadIdx.x & 31;
  const int tile = blockIdx.x * 4 + wave;
  const int ti = tile >> 5;        // row tile  (0..31)
  const int tj = tile & 31;        // col tile  (0..31)
  const int hi = lane >> 4;        // 0: lanes 0-15, 1: lanes 16-31
  const int l  = lane & 15;

  const float* rowA = X + (ti * 16 + l) * DIM;
  const float* rowB = X + (tj * 16 + l) * DIM;

  v8f acc = {};
#pragma unroll
  for (int k0 = 0; k0 < DIM; k0 += 4) {
    const int kk = k0 + 2 * hi;
    v2f a = *(const v2f*)(rowA + kk);
    v2f b = *(const v2f*)(rowB + kk);
    // 8 args: (neg_a, A, neg_b, B, c_mod, C, reuse_a, reuse_b)
    acc = __builtin_amdgcn_wmma_f32_16x16x4_f32(
        false, a, false, b, (short)0, acc, false, false);
  }

  float* out = G + (ti * 16 + 8 * hi) * NPTS + tj * 16 + l;
#pragma unroll
  for (int v = 0; v < 8; ++v) {
    out[v * NPTS] = acc[v];
  }
}

// ---------------------------------------------------------------------------
// Kernel 2: per-anchor triplet partial sums.
// Block i handles anchor i.  dist(i,j) = sqrt(max(G[i,i]+G[j,j]-2G[i,j], eps))
// staged in LDS; fixed-order accumulation + fixed-tree reduction => bitwise
// deterministic partials (no FP atomics).
// ---------------------------------------------------------------------------
__global__ __launch_bounds__(256) void triplet_rows_kernel(
    const float* __restrict__ G, const int* __restrict__ targets,
    float* __restrict__ partials) {
  __shared__ float drow[NPTS];
  __shared__ int   cls[NPTS];
  __shared__ float redn[256];
  __shared__ float redc[256];

  const int i = blockIdx.x;
  const int t = threadIdx.x;

  const float sqi = G[i * NPTS + i];
  for (int j = t; j < NPTS; j += 256) {
    const float sqj = G[j * NPTS + j];
    const float d2  = sqi + sqj - 2.0f * G[i * NPTS + j];
    drow[j] = sqrtf(fmaxf(d2, TLEPS));
    cls[j]  = targets[j];
  }
  __syncthreads();

  const int ci = cls[i];
  float num = 0.0f;
  float cnt = 0.0f;
  for (int j = 0; j < NPTS; ++j) {      // positives (diagonal included)
    if (cls[j] == ci) {
      const float dp = drow[j];
      for (int k = t; k < NPTS; k += 256) {  // negatives, strided per thread
        if (cls[k] != ci) {
          const float v = dp - drow[k] + TLMARGIN;
          num += fmaxf(v, 0.0f);
          cnt += 1.0f;
        }
      }
    }
  }

  redn[t] = num;
  redc[t] = cnt;
  __syncthreads();
  for (int s = 128; s > 0; s >>= 1) {
    if (t < s) {
      redn[t] += redn[t + s];
      redc[t] += redc[t + s];
    }
    __syncthreads();
  }
  if (t == 0) {
    partials[2 * i]     = redn[0];
    partials[2 * i + 1] = redc[0];
  }
}

// ---------------------------------------------------------------------------
// Kernel 3: deterministic final reduction, out = num / den.
// Total triplet count ~4.06e6 < 2^24, so the f32 count is exact.
// ---------------------------------------------------------------------------
__global__ __launch_bounds__(512) void triplet_finalize_kernel(
    const float* __restrict__ partials, float* __restrict__ out) {
  __shared__ float rn[NPTS];
  __shared__ float rd[NPTS];
  const int t = threadIdx.x;
  rn[t] = partials[2 * t];
  rd[t] = partials[2 * t + 1];
  __syncthreads();
  for (int s = 256; s > 0; s >>= 1) {
    if (t < s) {
      rn[t] += rn[t + s];
      rd[t] += rd[t + s];
    }
    __syncthreads();
  }
  if (t == 0) {
    out[0] = rn[0] / rd[0];
  }
}

extern "C" void kernel_launch(void* const* d_in, const int* in_sizes, int n_in,
                              void* d_out, int out_size, void* d_ws, size_t ws_size,
                              hipStream_t stream) {
  const float* X       = (const float*)d_in[0];   // [512,128] f32
  const int*   targets = (const int*)d_in[1];     // [512] int
  float* out = (float*)d_out;                     // scalar f32

  float* G        = (float*)d_ws;                 // 512*512 f32 = 1 MB
  float* partials = G + NPTS * NPTS;              // 512*2 f32

  // 1024 tiles of 16x16, 4 waves (128 threads) per block -> 256 blocks.
  gram_wmma_kernel<<<256, 128, 0, stream>>>(X, G);
  triplet_rows_kernel<<<NPTS, 256, 0, stream>>>(G, targets, partials);
  triplet_finalize_kernel<<<1, 512, 0, stream>>>(partials, out);
}